// ChildSumTreeLSTM_64690797412462
// MI455X (gfx1250) — compile-verified
//
#include <hip/hip_runtime.h>
#include <hip/hip_bf16.h>

// ChildSum TreeLSTM on MI455X (gfx1250, wave32).
//
// Stage 1 (leaf proj): h = x @ W_in + b_in via V_WMMA_F32_16X16X4_F32.
//   M=131072,K=768,N=128. HBM-bound: 402MB of x @ 23.3TB/s ~= 17us. fp32 WMMA
//   keeps exact reference numerics; matrix ceiling is irrelevant at this AI.
//   64-row M-tiles make the B operand stationary across 4 WMMAs (4x less
//   W_in L2 traffic: ~0.75GB) and the x stream is double-buffered through LDS
//   with GLOBAL_LOAD_ASYNC_TO_LDS_B128 (ASYNCcnt-tracked).
// Stage 2 (14 levels): gates = mean(h_children) @ W_up + b_up, LSTM gating.
//   Per-level h/c (<=64MB) and W_up (256KB) are L2-resident (192MB L2).
//   64 parents/block, each wave computes its 16 gate-local cols for ALL 4
//   gates -> gating entirely in accumulator registers, no LDS exchange.
//
// Workspace (d_ws) layout, needs 160MB:
//   [0,64MB) h ping (leaf)  [64,96MB) h pong  [96,128MB) c ping  [128,160MB) c pong

typedef float v2f __attribute__((ext_vector_type(2)));
typedef float v8f __attribute__((ext_vector_type(8)));
typedef int   vi4 __attribute__((vector_size(4 * sizeof(int))));

#define DH    128
#define DIN   768
#define GATES 512
#define PITCH 132          // 128 + 4: stride-128 A reads hit distinct banks; 16B aligned rows
#define MTILE 64           // rows per block = 4 sub-tiles of 16

#if __has_builtin(__builtin_amdgcn_global_load_async_to_lds_b128)
#define USE_ASYNC 1
typedef __attribute__((address_space(1))) vi4* g4_ptr;   // global int4*
typedef __attribute__((address_space(3))) vi4* l4_ptr;   // LDS int4*
#endif

__device__ __forceinline__ void wait_async_all() {
#if defined(USE_ASYNC)
#if __has_builtin(__builtin_amdgcn_s_wait_asynccnt)
    __builtin_amdgcn_s_wait_asynccnt(0);
#else
    asm volatile("s_wait_asynccnt 0x0" ::: "memory");
#endif
#endif
}

__device__ __forceinline__ float sigmoidf_fast(float x) {
    return 1.0f / (1.0f + __expf(-x));
}

// Stage one 64-row x 128-col fp32 chunk of x into an LDS buffer.
// Async path: per-lane b128 copies tracked by ASYNCcnt (no VGPR data).
__device__ __forceinline__ void stage_x_chunk(const float* __restrict__ x,
                                              size_t Rbase, int kc,
                                              float* buf, int tid) {
#pragma unroll
    for (int i = 0; i < 8; ++i) {
        const int e  = tid + 256 * i;          // 2048 float4 slots
        const int r  = e >> 5, c4 = e & 31;
        const float* g = x + (Rbase + r) * DIN + kc * 128 + c4 * 4;
        float*       l = buf + r * PITCH + c4 * 4;
#if defined(USE_ASYNC)
        __builtin_amdgcn_global_load_async_to_lds_b128((g4_ptr)g, (l4_ptr)l, 0, 0);
#else
        *reinterpret_cast<float4*>(l) = *reinterpret_cast<const float4*>(g);
#endif
    }
}

// ---------------------------------------------------------------------------
// Stage 1: leaf projection GEMM. Block = 64 rows x 128 cols, 8 waves.
// Wave w: cols [16w,16w+16), 4 row sub-tiles; B stationary across sub-tiles.
// ---------------------------------------------------------------------------
__global__ __launch_bounds__(256) void tlstm_leaf_proj(
    const float* __restrict__ x,     // [131072, 768]
    const float* __restrict__ Win,   // [768, 128]
    const float* __restrict__ bin,   // [128]
    float* __restrict__ h)           // [131072, 128]
{
    __shared__ float ldsX[2][MTILE * PITCH];   // 2 x 33.8KB double buffer

    const int tid  = threadIdx.x;
    const int lane = tid & 31;
    const int wave = tid >> 5;
    const int n    = lane & 15;
    const int koff = (lane >> 4) * 2;
    const int colbase = wave * 16;
    const size_t Rbase = (size_t)blockIdx.x * MTILE;

    v8f acc[4];
#pragma unroll
    for (int m = 0; m < 4; ++m) acc[m] = (v8f){0.f,0.f,0.f,0.f,0.f,0.f,0.f,0.f};

    stage_x_chunk(x, Rbase, 0, ldsX[0], tid);

    for (int kc = 0; kc < DIN / 128; ++kc) {
        const int bsel = kc & 1;
        wait_async_all();          // my chunk-kc copies have landed in LDS
        __syncthreads();           // everyone's landed; prior readers done
        if (kc + 1 < DIN / 128)
            stage_x_chunk(x, Rbase, kc + 1, ldsX[1 - bsel], tid);   // overlap

        const float* wk  = Win + (size_t)kc * 128 * DH;
        const float* lds = ldsX[bsel];
#pragma unroll 4
        for (int ks = 0; ks < 32; ++ks) {
            const int kk = ks * 4;
            v2f b;
            b[0] = wk[(size_t)(kk + koff) * DH + colbase + n];
            b[1] = wk[(size_t)(kk + koff + 1) * DH + colbase + n];
#pragma unroll
            for (int m = 0; m < 4; ++m) {
                v2f a;
                a[0] = lds[(16 * m + n) * PITCH + kk + koff];
                a[1] = lds[(16 * m + n) * PITCH + kk + koff + 1];
                acc[m] = __builtin_amdgcn_wmma_f32_16x16x4_f32(
                    false, a, false, b, (short)0, acc[m], false, false);
            }
        }
    }

    const float bias = bin[colbase + n];
#pragma unroll
    for (int m = 0; m < 4; ++m)
#pragma unroll
        for (int v = 0; v < 8; ++v) {
            const int row = 16 * m + v + (lane >> 4) * 8;
            h[(Rbase + row) * DH + colbase + n] = acc[m][v] + bias;
        }
}

// ---------------------------------------------------------------------------
// Stage 2: one tree level. Block = 64 parents x 512 gate cols, 8 waves.
// Wave w: gate-local cols [16w,16w+16) across all 4 gates x 4 row sub-tiles.
// ---------------------------------------------------------------------------
__global__ __launch_bounds__(256) void tlstm_level(
    const float* __restrict__ hC,    // [2*numParents, 128]
    const float* __restrict__ cC,    // [2*numParents, 128] (unused if leaf)
    const float* __restrict__ Wup,   // [128, 512]
    const float* __restrict__ bup,   // [512]
    float* __restrict__ hP,          // [numParents, 128]
    float* __restrict__ cP,          // [numParents, 128]
    int numParents, int leaf)
{
    __shared__ float ldsH[MTILE * PITCH];

    const int tid  = threadIdx.x;
    const int lane = tid & 31;
    const int wave = tid >> 5;
    const int n    = lane & 15;
    const int koff = (lane >> 4) * 2;
    const size_t Rbase = (size_t)blockIdx.x * MTILE;

    // h_avg = 0.5*(h[2P] + h[2P+1]) for 64 parents x 128 into LDS (zero-pad OOB).
#pragma unroll
    for (int i = 0; i < 8; ++i) {
        const int e = tid + 256 * i;
        const int r = e >> 5, c4 = e & 31;
        const size_t P = Rbase + r;
        float4 v = make_float4(0.f, 0.f, 0.f, 0.f);
        if (P < (size_t)numParents) {
            const float4 a = *reinterpret_cast<const float4*>(hC + (2 * P) * DH + c4 * 4);
            const float4 b = *reinterpret_cast<const float4*>(hC + (2 * P + 1) * DH + c4 * 4);
            v.x = 0.5f * (a.x + b.x);  v.y = 0.5f * (a.y + b.y);
            v.z = 0.5f * (a.z + b.z);  v.w = 0.5f * (a.w + b.w);
        }
        *reinterpret_cast<float4*>(&ldsH[r * PITCH + c4 * 4]) = v;
    }
    __syncthreads();

    v8f acc[4][4];   // [row sub-tile][gate]
#pragma unroll
    for (int m = 0; m < 4; ++m)
#pragma unroll
        for (int g = 0; g < 4; ++g) acc[m][g] = (v8f){0.f,0.f,0.f,0.f,0.f,0.f,0.f,0.f};

#pragma unroll 2
    for (int ks = 0; ks < 32; ++ks) {     // K = 128, 4 per WMMA
        const int kk = ks * 4;
        v2f a[4];
#pragma unroll
        for (int m = 0; m < 4; ++m) {
            a[m][0] = ldsH[(16 * m + n) * PITCH + kk + koff];
            a[m][1] = ldsH[(16 * m + n) * PITCH + kk + koff + 1];
        }
#pragma unroll
        for (int g = 0; g < 4; ++g) {
            v2f b;
            const int gc = g * DH + wave * 16 + n;
            b[0] = Wup[(size_t)(kk + koff) * GATES + gc];
            b[1] = Wup[(size_t)(kk + koff + 1) * GATES + gc];
#pragma unroll
            for (int m = 0; m < 4; ++m)
                acc[m][g] = __builtin_amdgcn_wmma_f32_16x16x4_f32(
                    false, a[m], false, b, (short)0, acc[m][g], false, false);
        }
    }

    // LSTM gating entirely in registers.
    const int gcol = wave * 16 + n;
    const float bi = bup[0 * DH + gcol];
    const float bo = bup[1 * DH + gcol];
    const float bu = bup[2 * DH + gcol];
    const float bf = bup[3 * DH + gcol];
#pragma unroll
    for (int m = 0; m < 4; ++m)
#pragma unroll
        for (int v = 0; v < 8; ++v) {
            const size_t P = Rbase + 16 * m + v + (lane >> 4) * 8;
            if (P < (size_t)numParents) {
                const float iv = sigmoidf_fast(acc[m][0][v] + bi);
                const float ov = sigmoidf_fast(acc[m][1][v] + bo);
                const float uv = tanhf(acc[m][2][v] + bu);
                const float fv = sigmoidf_fast(acc[m][3][v] + bf);
                float cs = 0.f;
                if (!leaf)
                    cs = cC[(2 * P) * DH + gcol] + cC[(2 * P + 1) * DH + gcol];
                const float cn = iv * uv + fv * cs;
                cP[P * DH + gcol] = cn;
                hP[P * DH + gcol] = ov * tanhf(cn);
            }
        }
}

// Emit root (h, c): out[0:1024) = h_root (8x128), out[1024:2048) = c_root.
__global__ void tlstm_emit(const float* __restrict__ h,
                           const float* __restrict__ c,
                           float* __restrict__ out)
{
    const int i = blockIdx.x * blockDim.x + threadIdx.x;
    if (i < 1024)       out[i] = h[i];
    else if (i < 2048)  out[i] = c[i - 1024];
}

extern "C" void kernel_launch(void* const* d_in, const int* in_sizes, int n_in,
                              void* d_out, int out_size, void* d_ws, size_t ws_size,
                              hipStream_t stream) {
    const float* x   = (const float*)d_in[0];   // [8,16384,768]
    const float* Win = (const float*)d_in[1];   // [768,128]
    const float* bin = (const float*)d_in[2];   // [128]
    const float* Wup = (const float*)d_in[3];   // [128,512]
    const float* bup = (const float*)d_in[4];   // [512]
    float* out = (float*)d_out;

    char* ws = (char*)d_ws;
    float* hb[2] = { (float*)ws,                   (float*)(ws + (64ull << 20)) };
    float* cb[2] = { (float*)(ws + (96ull << 20)), (float*)(ws + (128ull << 20)) };

    // Stage 1: 131072 rows / 64 per block.
    tlstm_leaf_proj<<<131072 / MTILE, 256, 0, stream>>>(x, Win, bin, hb[0]);

    // Stage 2: 14 levels, ping-pong h/c buffers.
    int s = 0;
    const float* hs = hb[0];
    const float* cs = nullptr;
    int n = 16384;
    int leaf = 1;
    while (n > 1) {
        const int parents = 8 * (n / 2);
        const int blocks  = (parents + MTILE - 1) / MTILE;
        float* hd = hb[1 - s];
        float* cd = cb[1 - s];
        tlstm_level<<<blocks, 256, 0, stream>>>(hs, cs, Wup, bup, hd, cd, parents, leaf);
        hs = hd; cs = cd;
        s = 1 - s;
        n >>= 1;
        leaf = 0;
    }

    tlstm_emit<<<8, 256, 0, stream>>>(hs, cs, out);
}